// KbModel_18116172054822
// MI455X (gfx1250) — compile-verified
//
#include <hip/hip_runtime.h>
#include <hip/hip_bf16.h>

#define C_X 640.0f
#define C_Y 480.0f
#define KB_EPS 1e-5f
#define NEWTON_MAX 100   // reference cap; early-exit once converged (same fixed point)

// Native clang vector: accepted by __builtin_nontemporal_{load,store},
// same memory layout as HIP float4, lowers to b128 global ops.
typedef float v4f __attribute__((ext_vector_type(4)));

// ---------------------------------------------------------------------------
// Per-point epilogue: given converged theta, ru, mx, my -> (u, v)
// ---------------------------------------------------------------------------
__device__ __forceinline__ void kb_project(
    float th, float ru, float mx, float my,
    float k0, float k1, float k2, float k3, float k4,
    float fx, float fy, float& u, float& v)
{
    float s = __sinf(th);                 // v_sin_f32
    float c = __cosf(th);                 // v_cos_f32
    float inv_r = 1.0f / (ru + KB_EPS);   // precise divide, once per point
    float x = s * mx * inv_r;
    float y = s * my * inv_r;
    float r2 = sqrtf(fmaf(x, x, y * y));
    float t2 = atan2f(r2, c);
    // d_theta = k0 + k1*t + k2*t^2 + k3*t^3 + k4*t^4  (Horner)
    float d = fmaf(fmaf(fmaf(fmaf(k4, t2, k3), t2, k2), t2, k1), t2, k0);
    u = fmaf(d * x, fx, C_X);
    v = fmaf(d * y, fy, C_Y);
}

// ---------------------------------------------------------------------------
// Main kernel: 4 points per thread (2x 128-bit NT loads, 2x 128-bit NT stores).
// 4 independent Newton chains per lane for ILP; wave32-uniform early exit.
// ---------------------------------------------------------------------------
__global__ __launch_bounds__(256) void kb_roundtrip_quad(
    const v4f* __restrict__ uv4,      // [2*nquads] v4f = two (u,v) pairs each
    const float* __restrict__ kvec,   // [5], uniform -> s_load
    const float* __restrict__ fxp,    // [1]
    const float* __restrict__ fyp,    // [1]
    v4f* __restrict__ out4,
    int nquads)
{
    const float k0 = kvec[0], k1 = kvec[1], k2 = kvec[2], k3 = kvec[3], k4 = kvec[4];
    // derivative coefficients: d/dt (k0 t + k1 t^2 + ... + k4 t^5)
    const float d1 = k0, d2 = 2.0f * k1, d3 = 3.0f * k2, d4 = 4.0f * k3, d5 = 5.0f * k4;
    const float fx = fxp[0], fy = fyp[0];
    const float inv_fx = 1.0f / fx, inv_fy = 1.0f / fy;

    int q = blockIdx.x * blockDim.x + threadIdx.x;
    if (q >= nquads) return;

    v4f A = __builtin_nontemporal_load(&uv4[2 * q + 0]); // (u0,v0,u1,v1)
    v4f B = __builtin_nontemporal_load(&uv4[2 * q + 1]); // (u2,v2,u3,v3)

    float mx[4] = { (A.x - C_X) * inv_fx, (A.z - C_X) * inv_fx,
                    (B.x - C_X) * inv_fx, (B.z - C_X) * inv_fx };
    float my[4] = { (A.y - C_Y) * inv_fy, (A.w - C_Y) * inv_fy,
                    (B.y - C_Y) * inv_fy, (B.w - C_Y) * inv_fy };
    float ru[4], th[4];
#pragma unroll
    for (int j = 0; j < 4; ++j) {
        ru[j] = sqrtf(fmaf(mx[j], mx[j], my[j] * my[j]));
        th[j] = ru[j];                       // Newton init: theta = r
    }

    // Newton solve of k0 t + k1 t^2 + k2 t^3 + k3 t^4 + k4 t^5 = ru.
    // Same fixed point as the reference's 100 fixed steps; exit as soon as the
    // whole wave (all 4 chains of all 32 lanes) has converged.
    for (int it = 0; it < NEWTON_MAX; ++it) {
        bool done = true;
#pragma unroll
        for (int j = 0; j < 4; ++j) {
            float t  = th[j];
            float p  = fmaf(fmaf(fmaf(fmaf(k4, t, k3), t, k2), t, k1), t, k0) * t;
            float dp = fmaf(fmaf(fmaf(fmaf(d5, t, d4), t, d3), t, d2), t, d1);
            // v_rcp_f32: ~1 ulp; Newton self-corrects, final theta is fp32-exact
            float delta = (p - ru[j]) * __builtin_amdgcn_rcpf(dp + KB_EPS);
            th[j] = t - delta;
            done = done && (__builtin_fabsf(delta) <=
                            fmaf(2.0e-7f, __builtin_fabsf(t), 1.0e-12f));
        }
        // wave32 vote -> uniform branch (s_cbranch), no divergence
        if (__builtin_amdgcn_ballot_w32(done) == __builtin_amdgcn_read_exec_lo())
            break;
    }

    float u[4], v[4];
#pragma unroll
    for (int j = 0; j < 4; ++j)
        kb_project(th[j], ru[j], mx[j], my[j], k0, k1, k2, k3, k4, fx, fy, u[j], v[j]);

    v4f O0 = { u[0], v[0], u[1], v[1] };
    v4f O1 = { u[2], v[2], u[3], v[3] };
    __builtin_nontemporal_store(O0, &out4[2 * q + 0]);
    __builtin_nontemporal_store(O1, &out4[2 * q + 1]);
}

// ---------------------------------------------------------------------------
// Scalar tail kernel (only launched if npts % 4 != 0; N=4M is divisible).
// ---------------------------------------------------------------------------
__global__ void kb_roundtrip_tail(
    const float* __restrict__ uv, const float* __restrict__ kvec,
    const float* __restrict__ fxp, const float* __restrict__ fyp,
    float* __restrict__ out, int start, int npts)
{
    int i = start + (int)(blockIdx.x * blockDim.x + threadIdx.x);
    if (i >= npts) return;

    const float k0 = kvec[0], k1 = kvec[1], k2 = kvec[2], k3 = kvec[3], k4 = kvec[4];
    const float d1 = k0, d2 = 2.0f * k1, d3 = 3.0f * k2, d4 = 4.0f * k3, d5 = 5.0f * k4;
    const float fx = fxp[0], fy = fyp[0];

    float mx = (uv[2 * i + 0] - C_X) / fx;
    float my = (uv[2 * i + 1] - C_Y) / fy;
    float ru = sqrtf(fmaf(mx, mx, my * my));
    float t = ru;
    for (int it = 0; it < NEWTON_MAX; ++it) {
        float p  = fmaf(fmaf(fmaf(fmaf(k4, t, k3), t, k2), t, k1), t, k0) * t;
        float dp = fmaf(fmaf(fmaf(fmaf(d5, t, d4), t, d3), t, d2), t, d1);
        float delta = (p - ru) / (dp + KB_EPS);
        t -= delta;
        if (__builtin_fabsf(delta) <= fmaf(2.0e-7f, __builtin_fabsf(t), 1.0e-12f)) break;
    }
    float u, v;
    kb_project(t, ru, mx, my, k0, k1, k2, k3, k4, fx, fy, u, v);
    out[2 * i + 0] = u;
    out[2 * i + 1] = v;
}

// ---------------------------------------------------------------------------
// Launch
// ---------------------------------------------------------------------------
extern "C" void kernel_launch(void* const* d_in, const int* in_sizes, int n_in,
                              void* d_out, int out_size, void* d_ws, size_t ws_size,
                              hipStream_t stream)
{
    (void)n_in; (void)out_size; (void)d_ws; (void)ws_size;
    const float* uv = (const float*)d_in[0];   // [N,2] fp32
    const float* kv = (const float*)d_in[1];   // [5]   fp32
    const float* fx = (const float*)d_in[2];   // [1]
    const float* fy = (const float*)d_in[3];   // [1]
    float* out = (float*)d_out;                // [N,2] fp32

    const int npts   = in_sizes[0] / 2;        // 4,194,304
    const int nquads = npts >> 2;

    if (nquads > 0) {
        const int block = 256;
        const int grid  = (nquads + block - 1) / block;
        kb_roundtrip_quad<<<grid, block, 0, stream>>>(
            (const v4f*)uv, kv, fx, fy, (v4f*)out, nquads);
    }
    const int tail = npts & 3;
    if (tail) {
        kb_roundtrip_tail<<<1, 64, 0, stream>>>(uv, kv, fx, fy, out, nquads << 2, npts);
    }
}